// ComplexGCN_55198919688425
// MI455X (gfx1250) — compile-verified
//
#include <hip/hip_runtime.h>

// ---------------------------------------------------------------------------
// 2-layer GCN (PyG GCNConv semantics) for MI455X / gfx1250.
//   h1   = x @ W1                         (WMMA f32 16x16x4, exact fp32)
//   agg1 = Dsym-normalized aggregate(h1)  (L2-resident gather + f32 atomics)
//   h2   = relu(agg1 + b1) @ W2           (WMMA, bias+relu fused into A load)
//   out  = Dsym-normalized aggregate(h2) + b2   -> d_out (complex view = same
//                                                 real storage, [N,128] f32)
// ---------------------------------------------------------------------------

#define N_NODES  50000
#define N_EDGES  800000
#define IN_DIM   128
#define HID      500
#define HID_PAD  512     // pad feature dim for clean 16-wide WMMA tiles
#define OUT_DIM  128

typedef __attribute__((ext_vector_type(2))) float v2f;
typedef __attribute__((ext_vector_type(8))) float v8f;

// ---------------- degree / normalization ----------------------------------

__global__ void k_deg_init(float* __restrict__ deg) {
  int i = blockIdx.x * 256 + threadIdx.x;
  if (i < N_NODES) deg[i] = 1.0f;                 // self-loop
}

__global__ void k_deg_count(const long long* __restrict__ ei,
                            float* __restrict__ deg) {
  int e = blockIdx.x * 256 + threadIdx.x;
  if (e < N_EDGES) {
    int d = (int)ei[N_EDGES + e];                 // dst row of edge_index
    atomicAdd(&deg[d], 1.0f);
  }
}

__global__ void k_rsqrt_inplace(float* __restrict__ deg) {
  int i = blockIdx.x * 256 + threadIdx.x;
  if (i < N_NODES) deg[i] = rsqrtf(deg[i]);       // deg >= 1 always
}

// ---------------- GEMM1: h1[50000 x 512] = x[50000 x 128] @ W1[128 x 500] --
// grid = (4, 3125), block = 256 (8 waves). Wave w of block bx owns output
// tile (tileM = 16*by, tileN = 16*(8*bx + w)). K loop: 32 x WMMA f32 16x16x4.
// A layout per lane: M = lane%16, K = k + 2*(lane/16) + {0,1}  (v2f, b64 load)
// B layout per lane: N = lane%16, K = k + 2*(lane/16) + {0,1}
// C/D: component r -> row tileM + r + 8*(lane/16), col tileN + lane%16.

__global__ void k_gemm1(const float* __restrict__ X,
                        const float* __restrict__ W1,
                        float* __restrict__ h1) {
  const int wave = threadIdx.x >> 5;
  const int lane = threadIdx.x & 31;
  const int half = lane >> 4;
  const int lm   = lane & 15;
  const int tileN = (blockIdx.x * 8 + wave) * 16;
  const int tileM = blockIdx.y * 16;
  const int n  = tileN + lm;
  const int nc = n < HID ? n : (HID - 1);         // clamp: no OOB speculation
  const bool nv = (n < HID);
  const float* __restrict__ arow = X + (size_t)(tileM + lm) * IN_DIM;

  v8f acc = {};
  for (int k = 0; k < IN_DIM; k += 4) {
    const int kk = k + 2 * half;
    v2f a = *(const v2f*)(arow + kk);             // 8B-aligned: kk is even
    v2f b;
    float b0 = W1[(size_t)kk * HID + nc];
    float b1v = W1[(size_t)(kk + 1) * HID + nc];
    b.x = nv ? b0  : 0.0f;                        // padded cols -> 0
    b.y = nv ? b1v : 0.0f;
    acc = __builtin_amdgcn_wmma_f32_16x16x4_f32(
        false, a, false, b, (short)0, acc, false, false);
  }
  float* __restrict__ orow = h1 + (size_t)(tileM + half * 8) * HID_PAD + n;
#pragma unroll
  for (int r = 0; r < 8; ++r) orow[(size_t)r * HID_PAD] = acc[r];
}

// ---------------- aggregation layer 1 --------------------------------------

// agg1[i] = dinv[i]^2 * h1[i]   (self-loop term; padded cols carry zeros)
__global__ void k_init_agg1(const float* __restrict__ h1,
                            const float* __restrict__ dinv,
                            float* __restrict__ agg1) {
  int idx = blockIdx.x * 256 + threadIdx.x;       // < 25.6M
  int i = idx >> 9;                               // / HID_PAD
  float di = dinv[i];
  agg1[idx] = di * di * h1[idx];
}

// one 256-thread block per edge: agg1[d] += dinv[s]*dinv[d] * h1[s]
__global__ void k_scatter1(const long long* __restrict__ ei,
                           const float* __restrict__ dinv,
                           const float* __restrict__ h1,
                           float* __restrict__ agg1) {
  const int e = blockIdx.x;
  const int s = (int)ei[e];
  const int d = (int)ei[N_EDGES + e];
  const float nm = dinv[s] * dinv[d];
  const float* __restrict__ hs = h1   + (size_t)s * HID_PAD;
  float* __restrict__       ad = agg1 + (size_t)d * HID_PAD;
  for (int c = threadIdx.x; c < HID; c += 256)
    atomicAdd(&ad[c], nm * hs[c]);
}

// ---------------- GEMM2: h2[50000 x 128] = relu(agg1+b1) @ W2[500 x 128] ---
// grid = 3125, block = 256; wave w owns col tile w (8*16 = 128 = OUT_DIM).
// K runs to exactly 500 (125 x WMMA), so padded agg1 columns are never read.

__global__ void k_gemm2(const float* __restrict__ agg1,
                        const float* __restrict__ B1,
                        const float* __restrict__ W2,
                        float* __restrict__ h2) {
  const int wave = threadIdx.x >> 5;
  const int lane = threadIdx.x & 31;
  const int half = lane >> 4;
  const int lm   = lane & 15;
  const int tileN = wave * 16;
  const int tileM = blockIdx.x * 16;
  const int n = tileN + lm;
  const float* __restrict__ arow = agg1 + (size_t)(tileM + lm) * HID_PAD;

  v8f acc = {};
  for (int k = 0; k < HID; k += 4) {
    const int kk = k + 2 * half;                  // max 498 -> kk+1 <= 499
    v2f a;
    a.x = fmaxf(arow[kk]     + B1[kk],     0.0f); // fused bias + ReLU
    a.y = fmaxf(arow[kk + 1] + B1[kk + 1], 0.0f);
    v2f b;
    b.x = W2[(size_t)kk * OUT_DIM + n];
    b.y = W2[(size_t)(kk + 1) * OUT_DIM + n];
    acc = __builtin_amdgcn_wmma_f32_16x16x4_f32(
        false, a, false, b, (short)0, acc, false, false);
  }
  float* __restrict__ orow = h2 + (size_t)(tileM + half * 8) * OUT_DIM + n;
#pragma unroll
  for (int r = 0; r < 8; ++r) orow[(size_t)r * OUT_DIM] = acc[r];
}

// ---------------- aggregation layer 2 (into d_out) -------------------------

// out[i] = dinv[i]^2 * h2[i] + b2   (self-loop + bias)
__global__ void k_init_out(const float* __restrict__ h2,
                           const float* __restrict__ dinv,
                           const float* __restrict__ b2,
                           float* __restrict__ out) {
  int idx = blockIdx.x * 256 + threadIdx.x;       // < 6.4M
  int i = idx >> 7;
  int c = idx & 127;
  float di = dinv[i];
  out[idx] = di * di * h2[idx] + b2[c];
}

// 128 threads share one edge: out[d] += dinv[s]*dinv[d] * h2[s]
__global__ void k_scatter2(const long long* __restrict__ ei,
                           const float* __restrict__ dinv,
                           const float* __restrict__ h2,
                           float* __restrict__ out) {
  int gid = blockIdx.x * 256 + threadIdx.x;       // < 102.4M
  int e = gid >> 7;
  int c = gid & 127;
  int s = (int)ei[e];
  int d = (int)ei[N_EDGES + e];
  float nm = dinv[s] * dinv[d];
  atomicAdd(&out[(size_t)d * OUT_DIM + c], nm * h2[(size_t)s * OUT_DIM + c]);
}

// ---------------------------------------------------------------------------

extern "C" void kernel_launch(void* const* d_in, const int* in_sizes, int n_in,
                              void* d_out, int out_size, void* d_ws, size_t ws_size,
                              hipStream_t stream) {
  const float*     x  = (const float*)d_in[0];
  const long long* ei = (const long long*)d_in[1];   // int64 edge_index [2, E]
  const float*     W1 = (const float*)d_in[2];
  const float*     B1 = (const float*)d_in[3];
  const float*     W2 = (const float*)d_in[4];
  const float*     b2 = (const float*)d_in[5];
  float* out = (float*)d_out;                        // [N,128] real storage

  // workspace layout (floats): dinv | h1 | agg1 | h2   (~231 MB total)
  float* dinv = (float*)d_ws;
  float* h1   = dinv + 65536;                        // keep 256B alignment
  float* agg1 = h1   + (size_t)N_NODES * HID_PAD;
  float* h2   = agg1 + (size_t)N_NODES * HID_PAD;

  // symmetric normalization: dinv = rsqrt(1 + in-degree)
  k_deg_init     <<<(N_NODES + 255) / 256, 256, 0, stream>>>(dinv);
  k_deg_count    <<<(N_EDGES + 255) / 256, 256, 0, stream>>>(ei, dinv);
  k_rsqrt_inplace<<<(N_NODES + 255) / 256, 256, 0, stream>>>(dinv);

  // layer 1
  k_gemm1    <<<dim3(HID_PAD / 128, N_NODES / 16), 256, 0, stream>>>(x, W1, h1);
  k_init_agg1<<<(N_NODES * HID_PAD) / 256, 256, 0, stream>>>(h1, dinv, agg1);
  k_scatter1 <<<N_EDGES, 256, 0, stream>>>(ei, dinv, h1, agg1);

  // layer 2 (bias + relu fused into GEMM2 A-operand)
  k_gemm2   <<<N_NODES / 16, 256, 0, stream>>>(agg1, B1, W2, h2);
  k_init_out<<<(N_NODES * OUT_DIM) / 256, 256, 0, stream>>>(h2, dinv, b2, out);
  k_scatter2<<<(N_EDGES * OUT_DIM) / 256, 256, 0, stream>>>(ei, dinv, h2, out);
}